// DualAttentionMultiRes_8804682957476
// MI455X (gfx1250) — compile-verified
//
#include <hip/hip_runtime.h>
#include <hip/hip_bf16.h>
#include <math.h>

// ---------------------------------------------------------------------------
// B=2, N=2048, QN=512, D=QD=1024, LD=64, H=QH=16, HD=QHD=64
// ---------------------------------------------------------------------------

typedef float v2f __attribute__((ext_vector_type(2)));
typedef float v4f __attribute__((ext_vector_type(4)));
typedef float v8f __attribute__((ext_vector_type(8)));
typedef int   v4i_gcc __attribute__((vector_size(16)));   // matches builtin's V4i

// ---- CDNA5 async global->LDS probe ----------------------------------------
#if defined(__has_builtin)
#if __has_builtin(__builtin_amdgcn_global_load_async_to_lds_b128) && \
    __has_builtin(__builtin_amdgcn_global_load_async_to_lds_b32)
#define USE_ASYNC_LDS 1
#endif
#if __has_builtin(__builtin_amdgcn_s_wait_asynccnt)
#define HAVE_WAIT_ASYNC_BUILTIN 1
#endif
#endif
#ifndef USE_ASYNC_LDS
#define USE_ASYNC_LDS 0
#endif

// Global (AS1) pointers share the generic representation; LDS (AS3) pointers
// are the low 32 bits of the generic pointer (aperture-based generic form).
#define AS1_P(T, p) ((__attribute__((address_space(1))) T*)(uintptr_t)(p))
#define AS3_P(T, p) ((__attribute__((address_space(3))) T*)(uint32_t)(uintptr_t)(p))

static __device__ __forceinline__ void async_copy_b128(const float* g, float* l) {
#if USE_ASYNC_LDS
  __builtin_amdgcn_global_load_async_to_lds_b128(AS1_P(v4i_gcc, g),
                                                 AS3_P(v4i_gcc, l), 0, 0);
#else
  (void)g; (void)l;
#endif
}
static __device__ __forceinline__ void async_copy_b32(const float* g, float* l) {
#if USE_ASYNC_LDS
  __builtin_amdgcn_global_load_async_to_lds_b32(AS1_P(int, g), AS3_P(int, l), 0, 0);
#else
  (void)g; (void)l;
#endif
}
static __device__ __forceinline__ void wait_async_zero() {
#if USE_ASYNC_LDS
#ifdef HAVE_WAIT_ASYNC_BUILTIN
  __builtin_amdgcn_s_wait_asynccnt(0);
#else
  asm volatile("s_wait_asynccnt 0x0" ::: "memory");
#endif
#endif
}

// CDNA5 native f32 matrix op: D(16x16) = A(16x4) * B(4x16) + C
static __device__ __forceinline__ v8f wmma4(v2f a, v2f b, v8f c) {
  return __builtin_amdgcn_wmma_f32_16x16x4_f32(false, a, false, b, (short)0, c,
                                               false, false);
}

// ---------------------------------------------------------------------------
// LDS-tiled, double-buffered f32 WMMA GEMM: C[M,Ntot] = A[M,K] @ W[K,Ntot]+bias
// Block = 256 threads (8 waves). Tile: BM=128, BN=64, KC=16. Grid(Ntot/64,M/128)
// LDS: A-panel [128][20] (row-major, pad 20 -> conflict-free, 8B-even frags),
//      W-panel transposed [64][20] so B-fragments are single ds_load_b64.
// Tiles staged with GLOBAL_LOAD_ASYNC_TO_LDS (chunk c+1 overlaps compute of c).
// ---------------------------------------------------------------------------
__global__ __launch_bounds__(256) void gemm_f32_wmma(
    const float* __restrict__ A, const float* __restrict__ W,
    const float* __restrict__ bias, float* __restrict__ C,
    int K, int Ntot)
{
  __shared__ float Ab[2][128][20];
  __shared__ float Bt[2][64][20];

  const int tid  = threadIdx.x;
  const int lane = tid & 31;
  const int wave = tid >> 5;
  const int l    = lane & 15;
  const int hh   = lane >> 4;
  const int m0   = blockIdx.y * 128;
  const int n0   = blockIdx.x * 64;

  // cooperative-load mapping
  const int arow = tid >> 1, aseg = (tid & 1) * 8;       // A: 8 floats/thread
  const int bkr  = tid >> 4, bc4  = (tid & 15) * 4;      // W: 4 floats/thread
  const float* agp = A + (size_t)(m0 + arow) * K + aseg;
  const float* bgp = W + (size_t)bkr * Ntot + n0 + bc4;

  v8f acc[4] = {};
  const int nchunk = K >> 4;

#if USE_ASYNC_LDS
  // ---- async DMA double-buffer pipeline ----
  async_copy_b128(agp,     &Ab[0][arow][aseg]);
  async_copy_b128(agp + 4, &Ab[0][arow][aseg + 4]);
#pragma unroll
  for (int i = 0; i < 4; ++i) async_copy_b32(bgp + i, &Bt[0][bc4 + i][bkr]);
  wait_async_zero();
  __syncthreads();

  for (int c = 0; c < nchunk; ++c) {
    const int buf = c & 1;
    if (c + 1 < nchunk) {            // kick chunk c+1 into the other buffer
      const float* ag = agp + (c + 1) * 16;
      const float* bg = bgp + (size_t)(c + 1) * 16 * Ntot;
      async_copy_b128(ag,     &Ab[buf ^ 1][arow][aseg]);
      async_copy_b128(ag + 4, &Ab[buf ^ 1][arow][aseg + 4]);
#pragma unroll
      for (int i = 0; i < 4; ++i) async_copy_b32(bg + i, &Bt[buf ^ 1][bc4 + i][bkr]);
    }
    const float* ar = &Ab[buf][wave * 16 + l][2 * hh];
#pragma unroll
    for (int k0 = 0; k0 < 16; k0 += 4) {
      v2f a = *(const v2f*)(ar + k0);
#pragma unroll
      for (int g = 0; g < 4; ++g) {
        v2f b = *(const v2f*)(&Bt[buf][g * 16 + l][k0 + 2 * hh]);
        acc[g] = wmma4(a, b, acc[g]);
      }
    }
    if (c + 1 < nchunk) wait_async_zero();
    __syncthreads();
  }
#else
  // ---- synchronous fallback: register-staged double buffer ----
  v4f ra0, ra1, rb;
  {
    ra0 = *(const v4f*)(agp);
    ra1 = *(const v4f*)(agp + 4);
    rb  = *(const v4f*)(bgp);
    *(v4f*)&Ab[0][arow][aseg]     = ra0;
    *(v4f*)&Ab[0][arow][aseg + 4] = ra1;
#pragma unroll
    for (int i = 0; i < 4; ++i) Bt[0][bc4 + i][bkr] = rb[i];
  }
  __syncthreads();
  for (int c = 0; c < nchunk; ++c) {
    const int buf = c & 1;
    if (c + 1 < nchunk) {
      const float* ag = agp + (c + 1) * 16;
      const float* bg = bgp + (size_t)(c + 1) * 16 * Ntot;
      ra0 = *(const v4f*)(ag);
      ra1 = *(const v4f*)(ag + 4);
      rb  = *(const v4f*)(bg);
    }
    const float* ar = &Ab[buf][wave * 16 + l][2 * hh];
#pragma unroll
    for (int k0 = 0; k0 < 16; k0 += 4) {
      v2f a = *(const v2f*)(ar + k0);
#pragma unroll
      for (int g = 0; g < 4; ++g) {
        v2f b = *(const v2f*)(&Bt[buf][g * 16 + l][k0 + 2 * hh]);
        acc[g] = wmma4(a, b, acc[g]);
      }
    }
    if (c + 1 < nchunk) {
      *(v4f*)&Ab[buf ^ 1][arow][aseg]     = ra0;
      *(v4f*)&Ab[buf ^ 1][arow][aseg + 4] = ra1;
#pragma unroll
      for (int i = 0; i < 4; ++i) Bt[buf ^ 1][bc4 + i][bkr] = rb[i];
    }
    __syncthreads();
  }
#endif

#pragma unroll
  for (int g = 0; g < 4; ++g) {
    const int   col = n0 + g * 16 + l;
    const float bv  = bias ? bias[col] : 0.0f;
#pragma unroll
    for (int j = 0; j < 8; ++j) {
      const size_t row = (size_t)m0 + wave * 16 + j + 8 * hh;
      C[row * (size_t)Ntot + col] = acc[g][j] + bv;
    }
  }
}

// ---------------------------------------------------------------------------
// One 16-key flash-attention step.  All K fragments are batch-loaded before
// the QK^T WMMA burst (two partial accumulators break the serial D->C chain);
// V fragments are prefetched before the softmax VALU so global latency hides
// under exp/shfl work.  P relayout C->A goes through a per-wave LDS tile
// (LDS ops from one wave complete in order - no barrier needed).
// ---------------------------------------------------------------------------
static __device__ __forceinline__ void flash_step(
    const v2f qf[16], const float* __restrict__ Kb, const float* __restrict__ Vb,
    size_t stride, int n0, int l, int hh, float (*P)[18],
    float m[8], float lsum[8], v8f o[4])
{
  // batched K fragments
  v2f kf[16];
  const float* kp = Kb + (size_t)(n0 + l) * stride + 2 * hh;
#pragma unroll
  for (int t = 0; t < 16; ++t) kf[t] = *(const v2f*)(kp + 4 * t);

  v8f s0 = {}, s1 = {};
#pragma unroll
  for (int t = 0; t < 8; ++t) {
    s0 = wmma4(qf[2 * t],     kf[2 * t],     s0);
    s1 = wmma4(qf[2 * t + 1], kf[2 * t + 1], s1);
  }
  v8f s = s0 + s1;

  // prefetch V fragments (consumed after softmax)
  v2f vf[16];
#pragma unroll
  for (int t = 0; t < 4; ++t) {
    const float* vp = Vb + (size_t)(n0 + 4 * t + 2 * hh) * stride + l;
#pragma unroll
    for (int g = 0; g < 4; ++g)
      vf[4 * t + g] = (v2f){vp[g * 16], vp[g * 16 + stride]};
  }

  // online softmax on the 16x16 score tile (C layout: reg j = rows j/j+8)
  float alpha[8];
#pragma unroll
  for (int j = 0; j < 8; ++j) {
    float rm = s[j];
    rm = fmaxf(rm, __shfl_xor(rm, 1, 32));
    rm = fmaxf(rm, __shfl_xor(rm, 2, 32));
    rm = fmaxf(rm, __shfl_xor(rm, 4, 32));
    rm = fmaxf(rm, __shfl_xor(rm, 8, 32));
    const float mn = fmaxf(m[j], rm);
    alpha[j] = __expf(m[j] - mn);
    const float p = __expf(s[j] - mn);
    float rs = p;
    rs += __shfl_xor(rs, 1, 32);
    rs += __shfl_xor(rs, 2, 32);
    rs += __shfl_xor(rs, 4, 32);
    rs += __shfl_xor(rs, 8, 32);
    lsum[j] = lsum[j] * alpha[j] + rs;
    m[j] = mn;
    P[j + 8 * hh][l] = p;
  }

#pragma unroll
  for (int g = 0; g < 4; ++g)
#pragma unroll
    for (int j = 0; j < 8; ++j) o[g][j] *= alpha[j];

  // P as A-fragments (batched from LDS), then 16 WMMAs over 4 chains
  v2f pa[4];
#pragma unroll
  for (int t = 0; t < 4; ++t) pa[t] = *(const v2f*)(&P[l][4 * t + 2 * hh]);
#pragma unroll
  for (int t = 0; t < 4; ++t)
#pragma unroll
    for (int g = 0; g < 4; ++g) o[g] = wmma4(pa[t], vf[4 * t + g], o[g]);
}

// ---------------------------------------------------------------------------
// Self-attention over x.  qkv:[B*N,3072] (Q@h*64, K@1024+h*64, V@2048+h*64).
// Output x_attn: [B*N, 1024].  Grid (N/64, H, B), 4 waves/block.
// ---------------------------------------------------------------------------
__global__ __launch_bounds__(128) void attn_self_kernel(
    const float* __restrict__ qkv, float* __restrict__ xattn)
{
  __shared__ float pt[4][16][18];
  const int lane = threadIdx.x & 31;
  const int wave = threadIdx.x >> 5;
  const int l    = lane & 15;
  const int hh   = lane >> 4;
  const int b = blockIdx.z, h = blockIdx.y;
  const int q0 = blockIdx.x * 64 + wave * 16;

  const float* base = qkv + (size_t)b * 2048 * 3072;
  const float* Qp = base + h * 64;
  const float* Kb = base + 1024 + h * 64;
  const float* Vb = base + 2048 + h * 64;

  v2f qf[16];
#pragma unroll
  for (int t = 0; t < 16; ++t) {
    v2f q = *(const v2f*)(Qp + (size_t)(q0 + l) * 3072 + 4 * t + 2 * hh);
    qf[t] = q * 0.125f;                 // fold 1/sqrt(64)
  }

  float m[8], lsum[8];
  v8f o[4] = {};
#pragma unroll
  for (int j = 0; j < 8; ++j) { m[j] = -1e30f; lsum[j] = 0.0f; }

  for (int n0 = 0; n0 < 2048; n0 += 16)
    flash_step(qf, Kb, Vb, 3072, n0, l, hh, pt[wave], m, lsum, o);

#pragma unroll
  for (int g = 0; g < 4; ++g)
#pragma unroll
    for (int j = 0; j < 8; ++j) {
      const size_t row = (size_t)b * 2048 + q0 + j + 8 * hh;
      xattn[row * 1024 + h * 64 + g * 16 + l] = o[g][j] / lsum[j];
    }
}

// ---------------------------------------------------------------------------
// Gated cross-attention (split softmax == two independent flash passes):
//   out = tanh(g_h)*softmax(Q K2^T)V2 + softmax(Q Kq^T)Vq
// ---------------------------------------------------------------------------
__global__ __launch_bounds__(128) void attn_cross_kernel(
    const float* __restrict__ xkv, const float* __restrict__ qqkv,
    const float* __restrict__ gate, float* __restrict__ qattn)
{
  __shared__ float pt[4][16][18];
  const int lane = threadIdx.x & 31;
  const int wave = threadIdx.x >> 5;
  const int l    = lane & 15;
  const int hh   = lane >> 4;
  const int b = blockIdx.z, h = blockIdx.y;
  const int q0 = blockIdx.x * 64 + wave * 16;

  const float* qbase = qqkv + (size_t)b * 512 * 3072;
  const float* Qp = qbase + h * 64;

  v2f qf[16];
#pragma unroll
  for (int t = 0; t < 16; ++t) {
    v2f q = *(const v2f*)(Qp + (size_t)(q0 + l) * 3072 + 4 * t + 2 * hh);
    qf[t] = q * 0.125f;
  }

  float m[8], lsum[8];
  v8f o[4] = {};
#pragma unroll
  for (int j = 0; j < 8; ++j) { m[j] = -1e30f; lsum[j] = 0.0f; }

  // phase 1: keys/values derived from x (xkv: [B*N,2048], k2@h*64, v2@1024+h*64)
  const float* kb1 = xkv + (size_t)b * 2048 * 2048 + h * 64;
  const float* vb1 = kb1 + 1024;
  for (int n0 = 0; n0 < 2048; n0 += 16)
    flash_step(qf, kb1, vb1, 2048, n0, l, hh, pt[wave], m, lsum, o);

  const float gh = tanhf(gate[h]);
  float o1[4][8];
#pragma unroll
  for (int g = 0; g < 4; ++g)
#pragma unroll
    for (int j = 0; j < 8; ++j) o1[g][j] = gh * o[g][j] / lsum[j];

  // phase 2: keys/values from the query stream (independent softmax)
  v8f vzero = {};
#pragma unroll
  for (int j = 0; j < 8; ++j) { m[j] = -1e30f; lsum[j] = 0.0f; }
#pragma unroll
  for (int g = 0; g < 4; ++g) o[g] = vzero;

  const float* kb2 = qbase + 1024 + h * 64;
  const float* vb2 = qbase + 2048 + h * 64;
  for (int n0 = 0; n0 < 512; n0 += 16)
    flash_step(qf, kb2, vb2, 3072, n0, l, hh, pt[wave], m, lsum, o);

#pragma unroll
  for (int g = 0; g < 4; ++g)
#pragma unroll
    for (int j = 0; j < 8; ++j) {
      const size_t row = (size_t)b * 512 + q0 + j + 8 * hh;
      qattn[row * 1024 + h * 64 + g * 16 + l] = o1[g][j] + o[g][j] / lsum[j];
    }
}

// ---------------------------------------------------------------------------
// low_res scramble + projection (tiny).
// A'[b,r,j] = low_res[b, (r*64+j)%256, (r*64+j)/256]; out = A' @ W + bias
// ---------------------------------------------------------------------------
__global__ __launch_bounds__(256) void lr_proj_kernel(
    const float* __restrict__ low_res, const float* __restrict__ Wl,
    const float* __restrict__ bl, float* __restrict__ out)
{
  const int idx = blockIdx.x * 256 + threadIdx.x;   // [0, 2*256*64)
  const int c = idx & 63;
  const int r = (idx >> 6) & 255;
  const int b = idx >> 14;
  const float* lr = low_res + (size_t)b * 256 * 64;
  float acc = bl[c];
#pragma unroll 8
  for (int j = 0; j < 64; ++j) {
    const int flat = r * 64 + j;
    acc += lr[(flat & 255) * 64 + (flat >> 8)] * Wl[j * 64 + c];
  }
  out[idx] = acc;
}

// ---------------------------------------------------------------------------
// Launch: 3 input GEMMs -> 2 attention kernels -> 2 output GEMMs + lr kernel.
// Workspace: qkv 12.58M | xkv 8.39M | qqkv 3.15M | xattn 4.19M | qattn 1.05M fl
// ---------------------------------------------------------------------------
extern "C" void kernel_launch(void* const* d_in, const int* in_sizes, int n_in,
                              void* d_out, int out_size, void* d_ws, size_t ws_size,
                              hipStream_t stream)
{
  (void)in_sizes; (void)n_in; (void)out_size; (void)ws_size;

  const float* x       = (const float*)d_in[0];
  const float* query   = (const float*)d_in[1];
  const float* low_res = (const float*)d_in[2];
  const float* W_qkv   = (const float*)d_in[3];
  const float* W_xkv   = (const float*)d_in[4];
  const float* W_qlin  = (const float*)d_in[5];
  const float* gate    = (const float*)d_in[6];
  const float* W_proj  = (const float*)d_in[7];
  const float* b_proj  = (const float*)d_in[8];
  const float* W_qproj = (const float*)d_in[9];
  const float* b_qproj = (const float*)d_in[10];
  const float* W_lr    = (const float*)d_in[11];
  const float* b_lr    = (const float*)d_in[12];

  float* out    = (float*)d_out;
  float* x_out  = out;                  // 2*2048*1024 = 4194304
  float* q_out  = out + 4194304;        // 2*512*1024  = 1048576
  float* lr_out = out + 5242880;        // 2*256*64    = 32768

  float* ws    = (float*)d_ws;
  float* qkv   = ws;                    // [4096, 3072]
  float* xkv   = qkv  + 12582912;       // [4096, 2048]
  float* qqkv  = xkv  + 8388608;        // [1024, 3072]
  float* xattn = qqkv + 3145728;        // [4096, 1024]
  float* qattn = xattn + 4194304;       // [1024, 1024]

  const dim3 gblk(256);

  // input projections
  gemm_f32_wmma<<<dim3(3072 / 64, 4096 / 128), gblk, 0, stream>>>(
      x, W_qkv, nullptr, qkv, 1024, 3072);
  gemm_f32_wmma<<<dim3(2048 / 64, 4096 / 128), gblk, 0, stream>>>(
      x, W_xkv, nullptr, xkv, 1024, 2048);
  gemm_f32_wmma<<<dim3(3072 / 64, 1024 / 128), gblk, 0, stream>>>(
      query, W_qlin, nullptr, qqkv, 1024, 3072);

  // attention
  attn_self_kernel<<<dim3(32, 16, 2), dim3(128), 0, stream>>>(qkv, xattn);
  attn_cross_kernel<<<dim3(8, 16, 2), dim3(128), 0, stream>>>(xkv, qqkv, gate, qattn);

  // output projections
  gemm_f32_wmma<<<dim3(1024 / 64, 4096 / 128), gblk, 0, stream>>>(
      xattn, W_proj, b_proj, x_out, 1024, 1024);
  gemm_f32_wmma<<<dim3(1024 / 64, 1024 / 128), gblk, 0, stream>>>(
      qattn, W_qproj, b_qproj, q_out, 1024, 1024);

  // low-res passthrough scramble + projection
  lr_proj_kernel<<<dim3(32768 / 256), dim3(256), 0, stream>>>(
      low_res, W_lr, b_lr, lr_out);
}